// vanilla_60790967108070
// MI455X (gfx1250) — compile-verified
//
#include <hip/hip_runtime.h>
#include <math.h>

// ---------------------------------------------------------------------------
// Complex RBF layer, fused for MI455X (gfx1250, wave32, WMMA).
//   v[b,n]  = ||x_b||^2 - 2*(X@G)[b,n] + ||G_:,n||^2          (GEMM1, f32 WMMA)
//   phi     = exp(-v / max(s, 1e-5))
//   y       = phi @ W (complex) + b                            (GEMM2, f32 WMMA)
// All matrix work on V_WMMA_F32_16X16X4_F32 to preserve f32 semantics.
// ---------------------------------------------------------------------------

typedef __attribute__((ext_vector_type(2))) float v2f;
typedef __attribute__((ext_vector_type(8))) float v8f;

static constexpr int Bn = 1024, In = 256, Nn = 1024, On = 256;
static constexpr float EPSv = 1e-5f;

#define WMMA_F32(a, b, c) \
  __builtin_amdgcn_wmma_f32_16x16x4_f32(false, (a), false, (b), (short)0, (c), false, false)

// ---- precompute gsq[n] = sum_i G[i,n]^2 (re & im) into workspace ----------
__global__ __launch_bounds__(256) void gsq_kernel(const float* __restrict__ Gre,
                                                  const float* __restrict__ Gim,
                                                  float* __restrict__ gsqre,
                                                  float* __restrict__ gsqim) {
  int n = blockIdx.x * 256 + threadIdx.x;
  float ar = 0.f, ai = 0.f;
  for (int i = 0; i < In; ++i) {
    float gr = Gre[i * Nn + n];
    float gi = Gim[i * Nn + n];
    ar += gr * gr;
    ai += gi * gi;
  }
  gsqre[n] = ar;
  gsqim[n] = ai;
}

// ---- fused: GEMM1 -> exp -> GEMM2 -> bias -> interleaved store ------------
__global__ __launch_bounds__(256) void rbf_fused(
    const float* __restrict__ xre, const float* __restrict__ xim,
    const float* __restrict__ Gre, const float* __restrict__ Gim,
    const float* __restrict__ sre, const float* __restrict__ sim,
    const float* __restrict__ Wre, const float* __restrict__ Wim,
    const float* __restrict__ bre, const float* __restrict__ bim,
    const float* __restrict__ gsqre, const float* __restrict__ gsqim,
    float* __restrict__ out) {
  // Padded LDS: row strides 258 / 130 floats keep ds_load_b64 A-fragments
  // conflict-free across the 64 LDS banks.
  __shared__ __align__(16) float Xre[16][258];
  __shared__ __align__(16) float Xim[16][258];
  __shared__ __align__(16) float Pre[16][130];
  __shared__ __align__(16) float Pim[16][130];
  __shared__ float xsre[16], xsim[16];

  const int tid  = threadIdx.x;
  const int wave = tid >> 5;
  const int lane = tid & 31;
  const int l16  = lane & 15;
  const int hi   = lane >> 4;   // 0: lanes 0-15, 1: lanes 16-31
  const int b0   = blockIdx.x * 16;

  // Stage the 16x256 X tile (re+im) into LDS, coalesced.
  for (int rep = 0; rep < 16; ++rep) {
    int flat = rep * 256 + tid;
    int row = flat >> 8, col = flat & 255;
    Xre[row][col] = xre[(b0 + row) * In + col];
    Xim[row][col] = xim[(b0 + row) * In + col];
  }
  __syncthreads();

  // ||x_b||^2 per row (16 re + 16 im) -- trivial cost.
  if (tid < 32) {
    int row = tid & 15;
    float s = 0.f;
    if (tid < 16) {
      for (int i = 0; i < In; ++i) { float v = Xre[row][i]; s += v * v; }
      xsre[row] = s;
    } else {
      for (int i = 0; i < In; ++i) { float v = Xim[row][i]; s += v * v; }
      xsim[row] = s;
    }
  }
  __syncthreads();

  // GEMM2 accumulators: 2 O-tiles x {re,im}, f32 16x16 each.
  v8f yr0 = {}, yr1 = {}, yi0 = {}, yi1 = {};

  for (int c = 0; c < 8; ++c) {       // 8 N-chunks of 128
    // ---------------- Phase 1: phi tile for this wave's 16 N-columns -------
    const int n0   = c * 128 + wave * 16;
    const int ncol = n0 + l16;        // this lane's N column (B/C/D col = lane%16)

    v8f accre = {}, accim = {};
    for (int k0 = 0; k0 < In; k0 += 4) {
      const int kk = k0 + 2 * hi;     // A/B f32 layout: lanes 16-31 hold K+2,K+3
      v2f are = *(const v2f*)&Xre[l16][kk];
      v2f aim = *(const v2f*)&Xim[l16][kk];
      v2f bgr, bgi;
      bgr.x = Gre[kk * Nn + ncol];
      bgr.y = Gre[(kk + 1) * Nn + ncol];
      bgi.x = Gim[kk * Nn + ncol];
      bgi.y = Gim[(kk + 1) * Nn + ncol];
      accre = WMMA_F32(are, bgr, accre);   // two independent WMMA chains
      accim = WMMA_F32(aim, bgi, accim);
    }

    const float invre = 1.0f / fmaxf(sre[ncol], EPSv);
    const float invim = 1.0f / fmaxf(sim[ncol], EPSv);
    const float gqr = gsqre[ncol], gqi = gsqim[ncol];
    const int loccol = wave * 16 + l16;
#pragma unroll
    for (int r = 0; r < 8; ++r) {
      const int m = hi * 8 + r;       // C/D layout: vgpr r -> row r (+8 for hi half)
      float vre = xsre[m] - 2.0f * accre[r] + gqr;
      float vim = xsim[m] - 2.0f * accim[r] + gqi;
      Pre[m][loccol] = __expf(-vre * invre);
      Pim[m][loccol] = __expf(-vim * invim);
    }
    __syncthreads();

    // ---------------- Phase 2: accumulate Y += phi_chunk @ W_chunk ---------
    const int o0 = wave * 32;
    for (int k0 = 0; k0 < 128; k0 += 4) {
      const int kk = k0 + 2 * hi;
      const int nk = c * 128 + kk;    // global K index into W rows
      v2f are = *(const v2f*)&Pre[l16][kk];
      v2f aim = *(const v2f*)&Pim[l16][kk];
      v2f aimn = -aim;                // f32 WMMA has no A/B NEG: negate data
      const float* wr = Wre + nk * On;
      const float* wi = Wim + nk * On;
      {
        const int o = o0 + l16;
        v2f bwr, bwi;
        bwr.x = wr[o];       bwr.y = wr[On + o];
        bwi.x = wi[o];       bwi.y = wi[On + o];
        yr0 = WMMA_F32(are,  bwr, yr0);   // + phi_re * W_re
        yr0 = WMMA_F32(aimn, bwi, yr0);   // - phi_im * W_im
        yi0 = WMMA_F32(are,  bwi, yi0);   // + phi_re * W_im
        yi0 = WMMA_F32(aim,  bwr, yi0);   // + phi_im * W_re
      }
      {
        const int o = o0 + 16 + l16;
        v2f bwr, bwi;
        bwr.x = wr[o];       bwr.y = wr[On + o];
        bwi.x = wi[o];       bwi.y = wi[On + o];
        yr1 = WMMA_F32(are,  bwr, yr1);
        yr1 = WMMA_F32(aimn, bwi, yr1);
        yi1 = WMMA_F32(are,  bwi, yi1);
        yi1 = WMMA_F32(aim,  bwr, yi1);
      }
    }
    __syncthreads();                  // phi buffer reused next chunk
  }

  // ---------------- Epilogue: bias + interleaved (re,im) store -------------
#pragma unroll
  for (int t = 0; t < 2; ++t) {
    const int o = wave * 32 + t * 16 + l16;
    const float br = bre[o], bi = bim[o];
    const v8f yr = t ? yr1 : yr0;
    const v8f yi = t ? yi1 : yi0;
#pragma unroll
    for (int r = 0; r < 8; ++r) {
      const int m = hi * 8 + r;
      v2f pack;
      pack.x = yr[r] + br;
      pack.y = yi[r] + bi;
      *(v2f*)(out + ((size_t)(b0 + m) * On + o) * 2) = pack;  // (B,1,O,2) layout
    }
  }
}

extern "C" void kernel_launch(void* const* d_in, const int* in_sizes, int n_in,
                              void* d_out, int out_size, void* d_ws, size_t ws_size,
                              hipStream_t stream) {
  const float* xre = (const float*)d_in[0];
  const float* xim = (const float*)d_in[1];
  const float* Gre = (const float*)d_in[2];
  const float* Gim = (const float*)d_in[3];
  const float* sre = (const float*)d_in[4];
  const float* sim = (const float*)d_in[5];
  const float* Wre = (const float*)d_in[6];
  const float* Wim = (const float*)d_in[7];
  const float* bre = (const float*)d_in[8];
  const float* bim = (const float*)d_in[9];
  float* out = (float*)d_out;

  float* gsqre = (float*)d_ws;          // 1024 floats
  float* gsqim = gsqre + Nn;            // 1024 floats (8 KB total scratch)

  gsq_kernel<<<dim3(Nn / 256), dim3(256), 0, stream>>>(Gre, Gim, gsqre, gsqim);
  rbf_fused<<<dim3(Bn / 16), dim3(256), 0, stream>>>(
      xre, xim, Gre, Gim, sre, sim, Wre, Wim, bre, bim, gsqre, gsqim, out);
}